// HOGLayerC_21036749815978
// MI455X (gfx1250) — compile-verified
//
#include <hip/hip_runtime.h>
#include <math.h>
#include <stdint.h>

// HOG layer, fully fused single pass:
//   x (32,3,224,224) f32  ->  out (32,27,28,28) f32
// grid = (14 strip rows, 96 planes), block = 256 threads (8 waves, wave32).
// Each block: async-copy (global -> LDS, ASYNCcnt) a reflect-padded 18x226
// tile (B128 interior + B32 reflected borders), then one wave per 8x8 cell:
// Sobel -> norm/bin -> 9-bin wave32 butterfly -> L2 normalize -> 9 outputs.

#define NBINS 9
#define POOL  8
#define HH    224
#define WW    224
#define HP    28                       // pooled H/W
#define RROWS 2                        // cell-rows per block
#define TROWS (RROWS * POOL + 2)       // 18 tile rows
#define TILE_W 226
#define TS     228                     // LDS row stride (floats); 228 % 64 = 36 -> conflict-free 3x3
#define CHUNKS_PER_ROW 56              // 224 / 4 (B128 chunks)
#define NCHUNK (TROWS * CHUNKS_PER_ROW)
#define PI_F   3.14159265358979323846f

__device__ __forceinline__ int reflect224(int i) {
    if (i < 0)   i = -i;               // -1 -> 1 (jnp 'reflect', no edge repeat)
    if (i > 223) i = 446 - i;          // 224 -> 222
    return i;
}

__global__ __launch_bounds__(256)
void hog_fused_kernel(const float* __restrict__ x, float* __restrict__ out) {
    const int rblk  = blockIdx.x;      // strip 0..13 (2 pooled rows each)
    const int plane = blockIdx.y;      // b*3 + c, 0..95
    const int tid   = threadIdx.x;
    const int lane  = tid & 31;
    const int wave  = tid >> 5;

    // +4 floats: room for the +3-float skew that 16B-aligns the interior (tx=1)
    __shared__ __align__(16) float tile[TROWS * TS + 4];
    float* tile0 = &tile[3];

    const float* src = x + (size_t)plane * (HH * WW);
    const int r0 = rblk * (RROWS * POOL);          // first input row of strip

    const unsigned base3 = (unsigned)(uintptr_t)(void*)&tile[0] + 12u; // LDS byte addr of tile0

    // ---- interior: async B128 global -> LDS (16B aligned both sides) ----
    for (int q = tid; q < NCHUNK; q += 256) {
        const int ty = q / CHUNKS_PER_ROW;
        const int j  = q - ty * CHUNKS_PER_ROW;    // 4-float chunk, gc = 4j
        const int gr = reflect224(r0 + ty - 1);
        const unsigned lds_addr = base3 + (unsigned)(ty * TS + 1 + 4 * j) * 4u;
        const uint64_t gaddr = (uint64_t)(uintptr_t)(src + (size_t)gr * WW + 4 * j);
        asm volatile("global_load_async_to_lds_b128 %0, %1, off"
                     :: "v"(lds_addr), "v"(gaddr) : "memory");
    }
    // ---- reflected border columns: async B32 (36 elements) ----
    if (tid < 2 * TROWS) {
        const int ty    = tid >> 1;
        const int right = tid & 1;
        const int gr = reflect224(r0 + ty - 1);
        const int tx = right ? (TILE_W - 1) : 0;   // tile col 225 or 0
        const int gc = right ? 222 : 1;            // reflect(224), reflect(-1)
        const unsigned lds_addr = base3 + (unsigned)(ty * TS + tx) * 4u;
        const uint64_t gaddr = (uint64_t)(uintptr_t)(src + (size_t)gr * WW + gc);
        asm volatile("global_load_async_to_lds_b32 %0, %1, off"
                     :: "v"(lds_addr), "v"(gaddr) : "memory");
    }
    asm volatile("s_wait_asynccnt 0" ::: "memory");
    __syncthreads();

    // ---- one wave per cell (2 rows x 28 cols = 56 cells, 7 per wave) ----
    for (int cc = wave; cc < RROWS * HP; cc += 8) {
        const int cy = (cc >= HP) ? 1 : 0;
        const int cx = cc - cy * HP;

        float n0 = 0.f, n1 = 0.f;
        int   b0 = 0,   b1 = 0;
        #pragma unroll
        for (int pp = 0; pp < 2; ++pp) {
            const int p  = lane + pp * 32;         // pixel 0..63 within 8x8 cell
            const int py = p >> 3;
            const int px = p & 7;
            const float* c = &tile0[(cy * POOL + py + 1) * TS + (cx * POOL + px + 1)];
            const float a00 = c[-TS - 1], a01 = c[-TS], a02 = c[-TS + 1];
            const float a10 = c[-1],                     a12 = c[+1];
            const float a20 = c[ TS - 1], a21 = c[ TS], a22 = c[ TS + 1];
            // cross-correlation, wx=[[1,0,-1],[2,0,-2],[1,0,-1]], wy = wx^T
            const float gx = (a00 - a02) + 2.f * (a10 - a12) + (a20 - a22);
            const float gy = (a00 - a20) + 2.f * (a01 - a21) + (a02 - a22);
            const float nn = sqrtf(gx * gx + gy * gy);
            const float ph = atan2f(gx, gy) * (9.0f / PI_F);
            int bb = (int)floorf(ph);
            bb %= NBINS; if (bb < 0) bb += NBINS;
            if (pp == 0) { n0 = nn; b0 = bb; } else { n1 = nn; b1 = bb; }
        }

        // 9-bin histogram: select + wave32 butterfly sum (all lanes get totals)
        float h[NBINS];
        #pragma unroll
        for (int k = 0; k < NBINS; ++k) {
            float v = (b0 == k ? n0 : 0.f) + (b1 == k ? n1 : 0.f);
            #pragma unroll
            for (int m = 16; m >= 1; m >>= 1)
                v += __shfl_xor(v, m, 32);
            h[k] = v;
        }

        float ss = 0.f;
        #pragma unroll
        for (int k = 0; k < NBINS; ++k) ss += h[k] * h[k];
        const float inv = 1.0f / fmaxf(sqrtf(ss), 1e-12f);

        if (lane < NBINS) {
            float val = h[0];
            #pragma unroll
            for (int k = 1; k < NBINS; ++k) val = (lane == k) ? h[k] : val;
            const int rout = rblk * RROWS + cy;    // pooled row 0..27
            // out[b, c*9+k, rout, cx] -> flat ((plane*9 + k)*28 + rout)*28 + cx
            out[(((size_t)plane * NBINS + lane) * HP + rout) * HP + cx] = val * inv;
        }
    }
}

extern "C" void kernel_launch(void* const* d_in, const int* in_sizes, int n_in,
                              void* d_out, int out_size, void* d_ws, size_t ws_size,
                              hipStream_t stream) {
    (void)in_sizes; (void)n_in; (void)out_size; (void)d_ws; (void)ws_size;
    const float* x = (const float*)d_in[0];   // (32,3,224,224) f32; Sobel weights are fixed constants
    float* out = (float*)d_out;               // (32,27,28,28) f32
    dim3 grid(HP / RROWS, 96);                // 14 strips x (32 batch * 3 chan) planes
    hipLaunchKernelGGL(hog_fused_kernel, grid, dim3(256), 0, stream, x, out);
}